// ModalitySampler_11184094839230
// MI455X (gfx1250) — compile-verified
//
#include <hip/hip_runtime.h>

// CDNA5 / gfx1250: wave32, WMMA 16x16x4 f32.
typedef __attribute__((ext_vector_type(2))) float v2f;
typedef __attribute__((ext_vector_type(8))) float v8f;

#define Hdim   512
#define Wdim   512
#define NT     6      // N_TARGETS
#define RAD    2      // RADIUS
#define RECL   5      // RECLEN
#define SVALID 507    // box-sum origins kept: 0 .. H-RECL-1

// One workgroup per image; 512 threads = 16 wave32.
__global__ __launch_bounds__(512)
void ModalitySampler_kernel(const float* __restrict__ hm,
                            int* __restrict__ out, int batch) {
    const int b = blockIdx.x;
    const float* img = hm + (size_t)b * Hdim * Wdim;

    __shared__ unsigned long long rowKey[512];  // (valBits<<32)|~col per S-row
    __shared__ unsigned int waveVal[16];
    __shared__ int          waveRow[16];
    __shared__ int zr[NT], zc[NT];              // zeroed 5x5 window origins
    __shared__ int bestRowSh, bestColSh;

    const int tid  = threadIdx.x;
    const int wave = tid >> 5;
    const int lane = tid & 31;
    const int m    = lane & 15;   // M row (A) / N col (B,D)
    const int half = lane >> 4;

    rowKey[tid] = 0ULL;
    __syncthreads();

    // ---------------- Phase A: bulk pass with WMMA --------------------------
    // S[i][j] = sum_{k=j..j+4} vert[i][k],  vert[i][k] = sum_{dr=0..4} hm[i+dr][k]
    // => S_tile = A(16x20 of vert) x Band(20x16), split into five K=4 WMMAs.
    for (int tile = wave; tile < 32 * 32; tile += 16) {
        const int rS0 = (tile >> 5) << 4;
        const int cS0 = (tile & 31) << 4;
        const int rr  = min(rS0 + m, SVALID - 1);     // clamp (masked later)
        v8f acc = {};
        #pragma unroll
        for (int kk = 0; kk < 5; ++kk) {
            // A operand layout (16x4 f32): lane m = M, element v -> K = 2*half+v
            const int kg0 = kk * 4 + half * 2;        // even
            const int kg1 = kg0 + 1;
            // Pair (kg0, kg1) is adjacent & 8B-aligned: one b64 load per row.
            const int ccp = min(cS0 + kg0, Wdim - 2); // clamp (masked later)
            const float* p = img + (size_t)rr * Wdim + ccp;
            v2f A = *(const v2f*)p;
            A += *(const v2f*)(p +     Wdim);
            A += *(const v2f*)(p + 2 * Wdim);
            A += *(const v2f*)(p + 3 * Wdim);
            A += *(const v2f*)(p + 4 * Wdim);
            // Banded B: B[k][j] = 1 iff j <= k <= j+4  (j = m in B layout)
            v2f B = {(m <= kg0 && kg0 <= m + 4) ? 1.0f : 0.0f,
                     (m <= kg1 && kg1 <= m + 4) ? 1.0f : 0.0f};
            acc = __builtin_amdgcn_wmma_f32_16x16x4_f32(
                false, A, false, B, (short)0, acc, false, false);
        }
        // D layout: acc[q] = S[rS0 + q + 8*half][cS0 + m]. Per-row argmax via
        // 16-lane xor shuffles (both halves reduce independently).
        #pragma unroll
        for (int q = 0; q < 8; ++q) {
            const int row  = rS0 + q + 8 * half;
            const int colg = cS0 + m;
            float val = (row < SVALID && colg < SVALID) ? acc[q] : -1.0f;
            unsigned int c = (unsigned int)colg;
            #pragma unroll
            for (int off = 1; off < 16; off <<= 1) {
                float        ov = __shfl_xor(val, off, 32);
                unsigned int oc = __shfl_xor(c,   off, 32);
                if (ov > val || (ov == val && oc < c)) { val = ov; c = oc; }
            }
            if (m == 0 && val >= 0.0f) {
                unsigned long long key =
                    ((unsigned long long)__float_as_uint(val) << 32) |
                    (unsigned int)(~c);
                atomicMax(&rowKey[row], key);   // LDS ds_max_u64
            }
        }
    }
    __syncthreads();

    // ---------------- Phase B: 6 NMS iterations -----------------------------
    for (int t = 0; t < NT; ++t) {
        // Global reduce over 507 row keys: max value, tie -> smallest row.
        int r = tid;
        unsigned long long key = (r < SVALID) ? rowKey[r] : 0ULL;
        unsigned int v = (unsigned int)(key >> 32);
        int br = r;
        #pragma unroll
        for (int off = 1; off < 32; off <<= 1) {
            unsigned int ov  = __shfl_xor(v,  off, 32);
            int          orr = __shfl_xor(br, off, 32);
            if (ov > v || (ov == v && orr < br)) { v = ov; br = orr; }
        }
        if (lane == 0) { waveVal[wave] = v; waveRow[wave] = br; }
        __syncthreads();
        if (tid == 0) {
            unsigned int bv = waveVal[0]; int bR = waveRow[0];
            for (int w2 = 1; w2 < 16; ++w2) {
                unsigned int wv = waveVal[w2]; int wr = waveRow[w2];
                if (wv > bv || (wv == bv && wr < bR)) { bv = wv; bR = wr; }
            }
            unsigned long long bk = rowKey[bR];
            int bC = (int)(~(unsigned int)bk);
            bestRowSh = bR; bestColSh = bC;
            zr[t] = bR; zc[t] = bC;
            // SWAP_RC: coords = [col + RADIUS, row + RADIUS]
            out[b * (NT * 2) + t * 2 + 0] = bC + RAD;
            out[b * (NT * 2) + t * 2 + 1] = bR + RAD;
        }
        __syncthreads();
        if (t == NT - 1) break;

        const int row = bestRowSh, col = bestColSh; (void)col;
        // Only S-rows row-4 .. row+4 change: reset + recompute them, masked.
        if (tid < 9) {
            int ra = row - 4 + tid;
            if (ra >= 0 && ra < SVALID) rowKey[ra] = 0ULL;
        }
        __syncthreads();
        for (int k = tid; k < 9 * 512; k += 512) {
            int ra = row - 4 + (k >> 9);
            int ca = k & 511;
            if (ra < 0 || ra >= SVALID || ca >= SVALID) continue;
            float s = 0.0f;
            #pragma unroll
            for (int dr2 = 0; dr2 < RECL; ++dr2) {
                int r2 = ra + dr2;
                #pragma unroll
                for (int dc2 = 0; dc2 < RECL; ++dc2) {
                    int c2 = ca + dc2;
                    float x = img[(size_t)r2 * Wdim + c2];
                    for (int w2 = 0; w2 <= t; ++w2) {
                        if (r2 >= zr[w2] && r2 < zr[w2] + RECL &&
                            c2 >= zc[w2] && c2 < zc[w2] + RECL) x = 0.0f;
                    }
                    s += x;
                }
            }
            unsigned long long key2 =
                ((unsigned long long)__float_as_uint(s) << 32) |
                (unsigned int)(~(unsigned int)ca);
            atomicMax(&rowKey[ra], key2);
        }
        __syncthreads();
    }
}

extern "C" void kernel_launch(void* const* d_in, const int* in_sizes, int n_in,
                              void* d_out, int out_size, void* d_ws, size_t ws_size,
                              hipStream_t stream) {
    (void)n_in; (void)out_size; (void)d_ws; (void)ws_size;
    const float* hm = (const float*)d_in[0];
    int* out = (int*)d_out;   // reference output dtype is int32
    int batch = in_sizes[0] / (Hdim * Wdim);
    ModalitySampler_kernel<<<batch, 512, 0, stream>>>(hm, out, batch);
}